// DiffTreeHyperLISTA_45449343926452
// MI455X (gfx1250) — compile-verified
//
#include <hip/hip_runtime.h>
#include <hip/hip_bf16.h>

typedef __attribute__((ext_vector_type(2))) float v2f;
typedef __attribute__((ext_vector_type(8))) float v8f;

#define NN 2048
#define MM 512
#define BB 512
#define LAYERS 16

// ---------------------------------------------------------------------------
// zero fill
__global__ void zero_k(float* p, int n) {
  int i = blockIdx.x * blockDim.x + threadIdx.x;
  if (i < n) p[i] = 0.f;
}

// ---------------------------------------------------------------------------
// tiled transpose: in[R,C] -> out[C,R]
__global__ __launch_bounds__(256) void transpose_k(const float* __restrict__ in,
                                                   float* __restrict__ out,
                                                   int R, int C) {
  __shared__ float tile[32][33];
  int bx = blockIdx.x * 32, by = blockIdx.y * 32;
  int tx = threadIdx.x, ty = threadIdx.y;  // 32 x 8
#pragma unroll
  for (int j = 0; j < 32; j += 8)
    tile[ty + j][tx] = in[(size_t)(by + ty + j) * C + (bx + tx)];
  __syncthreads();
#pragma unroll
  for (int j = 0; j < 32; j += 8)
    out[(size_t)(bx + ty + j) * R + (by + tx)] = tile[tx][ty + j];
}

// ---------------------------------------------------------------------------
// Generic WMMA f32 GEMM: OUT[BB x Ncols] = Add(+opt) + sgn * (X[BB x Kdim] * Mat[Kdim x Ncols])
// One wave computes a 16x64 strip: 4 accumulators sharing the A fragment.
// A frag (16x4 f32): lane<16 -> row=l16, K = 0/1 ; lane>=16 -> row=l16, K = 2/3
// B frag (4x16 f32): v.x -> K = 2*half, v.y -> K = 2*half+1, col = l16
// C/D  (16x16 f32): VGPR r -> row = r + 8*half, col = l16
__global__ __launch_bounds__(256)
void gemm_wmma_k(const float* __restrict__ X, const float* __restrict__ Mat,
                 const float* __restrict__ Add, float sgn,
                 float* __restrict__ OUT, int Kdim, int Ncols) {
  const int lane = threadIdx.x & 31;
  const int wave = threadIdx.x >> 5;
  const int l16  = lane & 15;
  const int half = lane >> 4;

  const int stripsPerRow = Ncols >> 6;
  int s    = blockIdx.x * 8 + wave;
  int brow = s / stripsPerRow;
  int scol = s - brow * stripsPerRow;
  int b0 = brow << 4;
  int n0 = scol << 6;

  v8f c0 = {}, c1 = {}, c2 = {}, c3 = {};
  const float* xrow  = X + (size_t)(b0 + l16) * Kdim + half * 2;
  const float* mbase = Mat + n0 + l16;

  for (int kk = 0; kk < Kdim; kk += 4) {
    v2f a = *(const v2f*)(xrow + kk);
    int kb = kk + half * 2;
    const float* m0 = mbase + (size_t)kb * Ncols;
    const float* m1 = mbase + (size_t)(kb + 1) * Ncols;
    v2f b0f, b1f, b2f, b3f;
    b0f.x = m0[0];  b0f.y = m1[0];
    b1f.x = m0[16]; b1f.y = m1[16];
    b2f.x = m0[32]; b2f.y = m1[32];
    b3f.x = m0[48]; b3f.y = m1[48];
    c0 = __builtin_amdgcn_wmma_f32_16x16x4_f32(false, a, false, b0f, (short)0, c0, false, false);
    c1 = __builtin_amdgcn_wmma_f32_16x16x4_f32(false, a, false, b1f, (short)0, c1, false, false);
    c2 = __builtin_amdgcn_wmma_f32_16x16x4_f32(false, a, false, b2f, (short)0, c2, false, false);
    c3 = __builtin_amdgcn_wmma_f32_16x16x4_f32(false, a, false, b3f, (short)0, c3, false, false);
  }

  v8f cc[4] = {c0, c1, c2, c3};
#pragma unroll
  for (int t = 0; t < 4; ++t) {
    int col = n0 + t * 16 + l16;
#pragma unroll
    for (int r = 0; r < 8; ++r) {
      int row = b0 + r + half * 8;
      size_t off = (size_t)row * Ncols + col;
      float v = sgn * cc[t][r];
      if (Add) v += Add[off];
      OUT[off] = v;
    }
  }
}

// ---------------------------------------------------------------------------
// Per-row: active count -> beta; z = x + beta*(x - x_prev) (or z = x on layer 0)
__global__ __launch_bounds__(256)
void rowstats_z_k(const float* __restrict__ x, const float* __restrict__ xprev,
                  const float* __restrict__ c2p, float* __restrict__ z, int first) {
  __shared__ float red[256];
  int b = blockIdx.x, tid = threadIdx.x;
  const float* xr  = x + (size_t)b * NN;
  const float* xpr = xprev + (size_t)b * NN;
  float xv[8];
  float cnt = 0.f;
#pragma unroll
  for (int j = 0; j < 8; ++j) {
    xv[j] = xr[tid + 256 * j];
    cnt += (fabsf(xv[j]) > 1e-6f) ? 1.f : 0.f;
  }
  red[tid] = cnt;
  __syncthreads();
  for (int o = 128; o > 0; o >>= 1) {
    if (tid < o) red[tid] += red[tid + o];
    __syncthreads();
  }
  float beta = (1.f / (1.f + expf(-c2p[0]))) * (red[0] / (float)NN);
  float* zr = z + (size_t)b * NN;
#pragma unroll
  for (int j = 0; j < 8; ++j) {
    int i = tid + 256 * j;
    zr[i] = first ? xv[j] : xv[j] + beta * (xv[j] - xpr[i]);
  }
}

// ---------------------------------------------------------------------------
// Per-row abs-sum of G; mode 0: store y_l1; mode 1: compute theta & K_tree
__global__ __launch_bounds__(256)
void rowabs_k(const float* __restrict__ G, float* __restrict__ yl1,
              float* __restrict__ theta, float* __restrict__ Ktree,
              const float* __restrict__ c1p, const float* __restrict__ c3p,
              float progress, int mode) {
  __shared__ float red[256];
  int b = blockIdx.x, tid = threadIdx.x;
  const float* g = G + (size_t)b * NN;
  float sm = 0.f;
#pragma unroll
  for (int j = 0; j < 8; ++j) sm += fabsf(g[tid + 256 * j]);
  red[tid] = sm;
  __syncthreads();
  for (int o = 128; o > 0; o >>= 1) {
    if (tid < o) red[tid] += red[tid + o];
    __syncthreads();
  }
  if (tid == 0) {
    float v = fmaxf(red[0], 1e-12f);
    if (mode == 0) {
      yl1[b] = v;
    } else {
      float yl = yl1[b];
      theta[b] = fabsf(c1p[0]) * fminf(fmaxf(v / yl, 0.f), 1.f);
      float lr  = fmaxf(logf(fmaxf(yl / v, 1.f)), 0.f);
      float sig = 1.f / (1.f + expf(-(lr - 1.f)));
      Ktree[b]  = fminf(fmaxf(fabsf(c3p[0]) * (float)NN * fmaxf(sig, progress), 1.f),
                        (float)NN * 0.6f);
    }
  }
}

// ---------------------------------------------------------------------------
// Batch means -> per-layer scalars: scal[0]=theta_mean (f32), ((int*)scal)[1]=k_idx
__global__ __launch_bounds__(512)
void scalars_k(const float* __restrict__ theta, const float* __restrict__ Ktree,
               float* __restrict__ scal) {
  __shared__ float st[512], sk[512];
  int tid = threadIdx.x;
  st[tid] = theta[tid];
  sk[tid] = Ktree[tid];
  __syncthreads();
  for (int o = 256; o > 0; o >>= 1) {
    if (tid < o) { st[tid] += st[tid + o]; sk[tid] += sk[tid + o]; }
    __syncthreads();
  }
  if (tid == 0) {
    scal[0] = st[0] * (1.f / (float)BB);
    int kidx = (int)floorf(sk[0] * (1.f / (float)BB));
    if (kidx < 1) kidx = 1;
    ((int*)scal)[1] = kidx;
  }
}

// ---------------------------------------------------------------------------
// Per-row: tree scores (heap, 11 levels) in LDS, exact k-th-largest via
// 4-round MSB radix select (scores >= 0 so float bits are order-preserving),
// then soft mask + soft threshold -> x_out.
__global__ __launch_bounds__(256)
void scores_update_k(const float* __restrict__ U, const float* __restrict__ scal,
                     float* __restrict__ Xout) {
  __shared__ float s[NN];
  __shared__ unsigned hist[256];
  __shared__ unsigned shp[2];  // prefix, remaining-k
  int b = blockIdx.x, tid = threadIdx.x;
  const float* urow = U + (size_t)b * NN;

  float uv[8], base[8];
#pragma unroll
  for (int j = 0; j < 8; ++j) {
    int i = tid + 256 * j;
    uv[j]   = urow[i];
    base[j] = fabsf(uv[j]);
    s[i]    = base[j];
  }
  __syncthreads();

  // level-by-level: score[i] = |u[i]| + 0.5 * score[parent]
  for (int d = 1; d <= 11; ++d) {
#pragma unroll
    for (int j = 0; j < 8; ++j) {
      int i = tid + 256 * j;
      int dep = 31 - __clz(i + 1);
      if (dep == d) s[i] = base[j] + 0.5f * s[(i - 1) >> 1];
    }
    __syncthreads();
  }

  int   kidx       = ((const int*)scal)[1];
  float theta_mean = scal[0];
  if (tid == 0) { shp[0] = 0u; shp[1] = (unsigned)kidx; }
  __syncthreads();

  for (int round = 0; round < 4; ++round) {
    hist[tid] = 0u;
    __syncthreads();
    int shift = 24 - 8 * round;
    unsigned pfx = shp[0];
#pragma unroll
    for (int j = 0; j < 8; ++j) {
      unsigned v = __float_as_uint(s[tid + 256 * j]);
      bool ok = (round == 0) || ((v >> (shift + 8)) == pfx);
      if (ok) atomicAdd(&hist[(v >> shift) & 255u], 1u);
    }
    __syncthreads();
    if (tid == 0) {
      unsigned kk = shp[1], cum = 0u, sel = 0u;
      for (int bb = 255; bb >= 0; --bb) {
        cum += hist[bb];
        if (cum >= kk) { sel = (unsigned)bb; kk -= (cum - hist[bb]); break; }
      }
      shp[0] = (pfx << 8) | sel;
      shp[1] = kk;
    }
    __syncthreads();
  }
  float tau = __uint_as_float(shp[0]);

  float* xrow = Xout + (size_t)b * NN;
#pragma unroll
  for (int j = 0; j < 8; ++j) {
    int i = tid + 256 * j;
    float mask = 1.f / (1.f + expf(-5.f * (s[i] - tau)));
    float mag  = fmaxf(base[j] - theta_mean, 0.f);
    float shr  = copysignf(mag, uv[j]);
    if (mag == 0.f) shr = 0.f;  // sign(0)*0 semantics
    xrow[i] = shr * mask;
  }
}

// ---------------------------------------------------------------------------
extern "C" void kernel_launch(void* const* d_in, const int* in_sizes, int n_in,
                              void* d_out, int out_size, void* d_ws, size_t ws_size,
                              hipStream_t stream) {
  const float* y  = (const float*)d_in[0];  // [B, M]
  const float* A  = (const float*)d_in[1];  // [M, N]
  const float* W  = (const float*)d_in[2];  // [M, N]
  const float* Ap = (const float*)d_in[3];  // [N, M]
  const float* c1 = (const float*)d_in[4];
  const float* c2 = (const float*)d_in[5];
  const float* c3 = (const float*)d_in[6];
  // parent/depth (d_in[7], d_in[8]) are recomputed analytically on device.

  const int BN = BB * NN;   // 1,048,576
  const int BM = BB * MM;   //   262,144
  float* ws  = (float*)d_ws;
  float* AT  = ws;          // A^T      [N, M]
  float* PT  = AT + BN;     // A_pinv^T [M, N]
  float* xA  = PT + BN;
  float* xB  = xA + BN;
  float* z   = xB + BN;
  float* Ub  = z + BN;
  float* G   = Ub + BN;
  float* R1  = G + BN;
  float* R2  = R1 + BM;
  float* yl1 = R2 + BM;
  float* th  = yl1 + BB;
  float* Kt  = th + BB;
  float* scal = Kt + BB;

  // init x / x_prev to zero (buffers are contiguous)
  zero_k<<<(2 * BN + 255) / 256, 256, 0, stream>>>(xA, 2 * BN);

  // one-time transposes for row-major inner-dim B operands
  transpose_k<<<dim3(NN / 32, MM / 32), dim3(32, 8), 0, stream>>>(A, AT, MM, NN);
  transpose_k<<<dim3(MM / 32, NN / 32), dim3(32, 8), 0, stream>>>(Ap, PT, NN, MM);

  // y_l1 = clip(sum |y @ A_pinv^T|), constant across layers
  gemm_wmma_k<<<128, 256, 0, stream>>>(y, PT, nullptr, 1.f, G, MM, NN);
  rowabs_k<<<BB, 256, 0, stream>>>(G, yl1, th, Kt, c1, c3, 0.f, 0);

  float* xc = xA;
  float* xp = xB;
  for (int k = 0; k < LAYERS; ++k) {
    rowstats_z_k<<<BB, 256, 0, stream>>>(xc, xp, c2, z, (k == 0) ? 1 : 0);
    // R1 = y - x @ A^T ; R2 = y - z @ A^T   (K = N = 2048, out cols = M = 512)
    gemm_wmma_k<<<32, 256, 0, stream>>>(xc, AT, y, -1.f, R1, NN, MM);
    gemm_wmma_k<<<32, 256, 0, stream>>>(z,  AT, y, -1.f, R2, NN, MM);
    // G = R1 @ A_pinv^T ; U = z + R2 @ W    (K = M = 512, out cols = N = 2048)
    gemm_wmma_k<<<128, 256, 0, stream>>>(R1, PT, nullptr, 1.f, G, MM, NN);
    gemm_wmma_k<<<128, 256, 0, stream>>>(R2, W,  z,       1.f, Ub, MM, NN);
    rowabs_k<<<BB, 256, 0, stream>>>(G, yl1, th, Kt, c1, c3, (float)(k + 1) / (float)LAYERS, 1);
    scalars_k<<<1, 512, 0, stream>>>(th, Kt, scal);
    scores_update_k<<<BB, 256, 0, stream>>>(Ub, scal, xp);
    float* t = xp; xp = xc; xc = t;
  }

  hipMemcpyAsync(d_out, xc, (size_t)BN * sizeof(float), hipMemcpyDeviceToDevice, stream);
}